// SparseConvNd_61993557951004
// MI455X (gfx1250) — compile-verified
//
#include <hip/hip_runtime.h>
#include <hip/hip_bf16.h>

// ---- problem constants (match reference setup_inputs) ----
constexpr int KSZ   = 3;
constexpr int RF    = 9;              // K^2
constexpr int CIN   = 64;
constexpr int COUT  = 64;
constexpr int WCOLS = COUT * RF;      // 576
constexpr int BB    = 2;
constexpr int HH    = 512;
constexpr int WW    = 512;
constexpr int GRID  = BB * HH * WW;   // 524288 = 2^19
constexpr int SCAN_BLK_ELEMS = 2048;  // per-block scan width
constexpr int SCAN_NBLOCKS   = GRID / SCAN_BLK_ELEMS; // 256
constexpr int PTS_PER_BLK    = 32;    // two 16-row WMMA tiles share each B fragment

typedef __attribute__((ext_vector_type(2))) float v2f;
typedef __attribute__((ext_vector_type(8))) float v8f;

__device__ __forceinline__ int cell_of(int b, int y, int x, int o) {
    int oy = o / KSZ, ox = o % KSZ;
    int yy = (y + oy) >> 1;
    int xx = (x + ox) >> 1;
    return (b * HH + yy) * WW + xx;
}

// 1) zero occupancy marks
__global__ void k_zero(unsigned* marks) {
    int i = blockIdx.x * blockDim.x + threadIdx.x;
    if (i < GRID) marks[i] = 0u;
}

// 2) mark occupied cells (benign write races)
__global__ void k_mark(const int* __restrict__ pts, unsigned* __restrict__ marks, int N) {
    int m = blockIdx.x * blockDim.x + threadIdx.x;
    if (m >= N * RF) return;
    int n = m / RF, o = m - n * RF;
    int b = pts[n * 3 + 0], y = pts[n * 3 + 1], x = pts[n * 3 + 2];
    marks[cell_of(b, y, x, o)] = 1u;
}

// 3) per-block exclusive scan over marks -> ranks (block-local) + block sums
__global__ __launch_bounds__(256) void k_scan1(const unsigned* __restrict__ marks,
                                               unsigned* __restrict__ ranks,
                                               unsigned* __restrict__ bsums) {
    __shared__ unsigned s[256];
    int tid  = threadIdx.x;
    int base = blockIdx.x * SCAN_BLK_ELEMS + tid * 8;
    unsigned v[8], run = 0u;
#pragma unroll
    for (int i = 0; i < 8; ++i) v[i] = marks[base + i];
#pragma unroll
    for (int i = 0; i < 8; ++i) { unsigned t = v[i]; v[i] = run; run += t; }
    s[tid] = run;
    __syncthreads();
    for (int off = 1; off < 256; off <<= 1) {
        unsigned t = (tid >= off) ? s[tid - off] : 0u;
        __syncthreads();
        s[tid] += t;
        __syncthreads();
    }
    unsigned excl = (tid == 0) ? 0u : s[tid - 1];
#pragma unroll
    for (int i = 0; i < 8; ++i) ranks[base + i] = v[i] + excl;
    if (tid == 255) bsums[blockIdx.x] = s[255];
}

// 4) exclusive scan of the 256 block sums (single block)
__global__ __launch_bounds__(256) void k_scan2(unsigned* bsums) {
    __shared__ unsigned s[256];
    int tid = threadIdx.x;
    s[tid] = bsums[tid];
    __syncthreads();
    for (int off = 1; off < 256; off <<= 1) {
        unsigned t = (tid >= off) ? s[tid - off] : 0u;
        __syncthreads();
        s[tid] += t;
        __syncthreads();
    }
    bsums[tid] = (tid == 0) ? 0u : s[tid - 1];
}

// 5) add scanned block offsets -> global sorted ranks
__global__ void k_addoff(unsigned* __restrict__ ranks, const unsigned* __restrict__ bsums) {
    int i = blockIdx.x * blockDim.x + threadIdx.x;
    if (i < GRID) ranks[i] += bsums[i >> 11];   // 2^11 = SCAN_BLK_ELEMS
}

// 6a) init outputs, vectorized: y rows = bias (b128 stores), idx rows = (0,0,0)
__global__ void k_init_v4(const float4* __restrict__ bias4, float4* __restrict__ y4,
                          float* __restrict__ idx_out, int M) {
    int t = blockIdx.x * blockDim.x + threadIdx.x;   // M * 16 threads
    if (t >= M * (COUT / 4)) return;
    int q = t & (COUT / 4 - 1);
    y4[t] = bias4[q];
    if (q == 0) {
        float* p = idx_out + (size_t)(t >> 4) * 3;
        p[0] = 0.0f; p[1] = 0.0f; p[2] = 0.0f;
    }
}

// 6b) scalar fallback (unaligned y base)
__global__ void k_init_s(const float* __restrict__ bias, float* __restrict__ y,
                         float* __restrict__ idx_out, int M) {
    int t = blockIdx.x * blockDim.x + threadIdx.x;
    if (t >= M * COUT) return;
    int c = t & (COUT - 1);
    y[t] = bias[c];
    if (c < 3) idx_out[(size_t)(t >> 6) * 3 + c] = 0.0f;
}

// 7) emit decoded (b,h,w) for occupied cells at their sorted rank
__global__ void k_emit(const unsigned* __restrict__ marks, const unsigned* __restrict__ ranks,
                       float* __restrict__ idx_out) {
    int cell = blockIdx.x * blockDim.x + threadIdx.x;
    if (cell >= GRID || !marks[cell]) return;
    size_t r = ranks[cell];
    int w = cell % WW;
    int rem = cell / WW;
    int h = rem % HH;
    int b = (rem / HH) % BB;
    idx_out[r * 3 + 0] = (float)b;
    idx_out[r * 3 + 1] = (float)h;
    idx_out[r * 3 + 2] = (float)w;
}

// 8) fused GEMM (V_WMMA_F32_16X16X4_F32) + atomic scatter.
//    288 threads = 9 waves; wave o owns RF-offset o. 32 points per block:
//    two 16-row A tiles share each B fragment (halves weight traffic,
//    8 WMMAs per K-step per wave).
__global__ __launch_bounds__(288) void k_gemm_scatter(const int* __restrict__ pts,
                                                      const float* __restrict__ vals,
                                                      const float* __restrict__ wts,
                                                      const unsigned* __restrict__ ranks,
                                                      float* __restrict__ y, int N) {
    __shared__ float As[PTS_PER_BLK][CIN + 1];   // +1 pad: conflict-free column reads
    __shared__ unsigned targ[RF * PTS_PER_BLK];  // scatter row per (offset, point)

    const int tid  = threadIdx.x;                // 0..287
    const int wave = tid >> 5;                   // 0..8 == RF offset
    const int lane = tid & 31;
    const int p0   = blockIdx.x * PTS_PER_BLK;

    // stage A tiles (32 x 64 fp32) into LDS
    for (int i = tid; i < PTS_PER_BLK * CIN; i += 288) {
        int p = p0 + (i >> 6);
        As[i >> 6][i & 63] = (p < N) ? vals[(size_t)p * CIN + (i & 63)] : 0.0f;
    }
    // one scatter target per thread: 288 == RF * 32
    {
        int o = tid >> 5, p = p0 + (tid & 31);
        unsigned r = 0xFFFFFFFFu;
        if (p < N) {
            int b = pts[p * 3 + 0], yy = pts[p * 3 + 1], xx = pts[p * 3 + 2];
            r = ranks[cell_of(b, yy, xx, o)];
        }
        targ[(o << 5) + (tid & 31)] = r;
    }
    __syncthreads();

    const int o    = wave;
    const int half = lane >> 4;                  // K-half select per ISA A/B layout
    const int l15  = lane & 15;

    // hoist this lane's 16 scatter targets (8 rows x 2 tiles) into registers
    unsigned rt[2][8];
#pragma unroll
    for (int tt = 0; tt < 2; ++tt)
#pragma unroll
        for (int j = 0; j < 8; ++j)
            rt[tt][j] = targ[(o << 5) + tt * 16 + 8 * half + j];

    v8f acc[2][4] = {};                          // two 16x64 fp32 tiles

    for (int k0 = 0; k0 < CIN; k0 += 4) {
        const int ka = k0 + 2 * half;
        v2f a0, a1;
        a0.x = As[l15][ka];      a0.y = As[l15][ka + 1];
        a1.x = As[16 + l15][ka]; a1.y = As[16 + l15][ka + 1];
#pragma unroll
        for (int ct = 0; ct < 4; ++ct) {
            const int gcol = o * COUT + ct * 16 + l15;
            v2f b;
            b.x = wts[(size_t)ka * WCOLS + gcol];
            b.y = wts[(size_t)(ka + 1) * WCOLS + gcol];
            acc[0][ct] = __builtin_amdgcn_wmma_f32_16x16x4_f32(
                false, a0, false, b, (short)0, acc[0][ct], false, false);
            acc[1][ct] = __builtin_amdgcn_wmma_f32_16x16x4_f32(
                false, a1, false, b, (short)0, acc[1][ct], false, false);
        }
    }

    // scatter: one guard per row, 4 coalesced f32 atomics per row
#pragma unroll
    for (int tt = 0; tt < 2; ++tt) {
#pragma unroll
        for (int j = 0; j < 8; ++j) {
            unsigned r = rt[tt][j];
            if (r != 0xFFFFFFFFu) {
                float* dst = &y[(size_t)r * COUT + l15];
                atomicAdd(dst +  0, acc[tt][0][j]);
                atomicAdd(dst + 16, acc[tt][1][j]);
                atomicAdd(dst + 32, acc[tt][2][j]);
                atomicAdd(dst + 48, acc[tt][3][j]);
            }
        }
    }
}

extern "C" void kernel_launch(void* const* d_in, const int* in_sizes, int n_in,
                              void* d_out, int out_size, void* d_ws, size_t ws_size,
                              hipStream_t stream) {
    const int*   pts  = (const int*)d_in[0];
    const float* vals = (const float*)d_in[1];
    const float* wts  = (const float*)d_in[2];
    const float* bias = (const float*)d_in[3];
    const int N = in_sizes[0] / 3;
    const int M = N * RF;

    unsigned* marks = (unsigned*)d_ws;                 // GRID u32
    unsigned* ranks = marks + GRID;                    // GRID u32
    unsigned* bsums = ranks + GRID;                    // 256 u32

    float* idx_out = (float*)d_out;                    // M*3 (int values as float)
    float* y       = idx_out + (size_t)M * 3;          // M*COUT

    k_zero<<<(GRID + 255) / 256, 256, 0, stream>>>(marks);
    k_mark<<<(M + 255) / 256, 256, 0, stream>>>(pts, marks, N);
    k_scan1<<<SCAN_NBLOCKS, 256, 0, stream>>>(marks, ranks, bsums);
    k_scan2<<<1, 256, 0, stream>>>(bsums);
    k_addoff<<<(GRID + 255) / 256, 256, 0, stream>>>(ranks, bsums);

    if ((((uintptr_t)y) & 15u) == 0) {
        k_init_v4<<<(M * (COUT / 4) + 255) / 256, 256, 0, stream>>>(
            (const float4*)bias, (float4*)y, idx_out, M);
    } else {
        k_init_s<<<(M * COUT + 255) / 256, 256, 0, stream>>>(bias, y, idx_out, M);
    }

    k_emit<<<(GRID + 255) / 256, 256, 0, stream>>>(marks, ranks, idx_out);
    k_gemm_scatter<<<(N + PTS_PER_BLK - 1) / PTS_PER_BLK, 288, 0, stream>>>(
        pts, vals, wts, ranks, y, N);
}